// CrossViewSparseSampler_5299989643278
// MI455X (gfx1250) — compile-verified
//
#include <hip/hip_runtime.h>

typedef __attribute__((ext_vector_type(2))) float v2f;
typedef __attribute__((ext_vector_type(8))) float v8f;

namespace {
constexpr int B_  = 4;
constexpr int V_  = 6;
constexpr int C_  = 256;
constexpr int Q_  = 2048;
constexpr int NS  = 24;              // V * KP slots per query
constexpr int W0_ = 176, H0_ = 64, HW0_ = W0_ * H0_;
constexpr int W1_ = 88,  H1_ = 32, HW1_ = W1_ * H1_;
}

__global__ __launch_bounds__(256)
void cvss_kernel(const float* __restrict__ feat0,
                 const float* __restrict__ feat1,
                 const float* __restrict__ refs,
                 const float* __restrict__ intr,
                 const float* __restrict__ extr,
                 float* __restrict__ out)
{
    __shared__ float s_tile[16 * 16];        // WMMA D spill for slot extraction
    __shared__ int   s_idx[2][NS][4];        // per-level per-slot corner indices
    __shared__ float s_w[2][NS][4];          // per-level per-slot corner weights (valid+inb folded)
    __shared__ float s_valid[NS];
    __shared__ float s_scale;                // 0.5 / max(sum(valid), 1)

    const int bq = blockIdx.x;
    const int b  = bq >> 11;                 // / Q_
    const int q  = bq & (Q_ - 1);
    const int t  = threadIdx.x;

    if (t < 32) {                            // wave 0: projection via WMMA
        const int  L  = t;
        const int  r  = L & 15;
        const bool hi = (L >= 16);

        const float* rp = refs + (size_t)(b * Q_ + q) * 3;
        const float rx = rp[0], ry = rp[1], rz = rp[2];

        // ---- B fragment: B[k][n] = homo component k of keypoint n (n<4), else 0.
        // Layout: VGPR0 = rows K0 (lanes 0-15) / K2 (lanes 16-31); VGPR1 = K1 / K3.
        v2f bfrag; bfrag[0] = 0.0f; bfrag[1] = 0.0f;
        if (r < 4) {
            if (!hi) {
                const float kx = (r == 1) ? 2.0f : ((r == 3) ? -2.0f : 0.0f);
                const float ky = (r == 2) ? 2.0f : 0.0f;
                bfrag[0] = rx + kx;          // K=0 row
                bfrag[1] = ry + ky;          // K=1 row
            } else {
                bfrag[0] = rz;               // K=2 row
                bfrag[1] = 1.0f;             // K=3 row
            }
        }

        // ---- A fragment: row = 3*viewLocal + component; A[row][k] = (K·E)[comp][k]
        // Layout: lane holds row (L&15); VGPR0 = cols K0/K2, VGPR1 = cols K1/K3.
        auto makeA = [&](int vbase, int nrows) -> v2f {
            v2f a; a[0] = 0.0f; a[1] = 0.0f;
            if (r < nrows) {
                const int v  = vbase + r / 3;
                const int i  = r % 3;
                const int j0 = hi ? 2 : 0;
                const float* Kp = intr + (size_t)(b * V_ + v) * 9 + i * 3;
                const float* Ep = extr + (size_t)(b * V_ + v) * 16;
                const float k0 = Kp[0], k1 = Kp[1], k2 = Kp[2];
                a[0] = k0 * Ep[0 * 4 + j0]     + k1 * Ep[1 * 4 + j0]     + k2 * Ep[2 * 4 + j0];
                a[1] = k0 * Ep[0 * 4 + j0 + 1] + k1 * Ep[1 * 4 + j0 + 1] + k2 * Ep[2 * 4 + j0 + 1];
            }
            return a;
        };

        // ---- per-slot record builder (both levels)
        auto record = [&](int slot, int v, float u, float vv, float w) {
            const float valid = (w > 0.0f) ? 1.0f : 0.0f;
            s_valid[slot] = valid;
            const float dz = fmaxf(w, 1e-5f);
            const float x = u / dz;
            const float y = vv / dz;
            #pragma unroll
            for (int l = 0; l < 2; ++l) {
                const int Wl  = l ? W1_  : W0_;
                const int Hl  = l ? H1_  : H0_;
                const int HWl = l ? HW1_ : HW0_;
                // clamp so int conversion is safe; exact for every in-range case
                const float xs = fminf(fmaxf(x, -8.0f), (float)(Wl + 8));
                const float ys = fminf(fmaxf(y, -8.0f), (float)(Hl + 8));
                const float x0f = floorf(xs), y0f = floorf(ys);
                const int x0 = (int)x0f, y0 = (int)y0f;
                const int x1 = x0 + 1,  y1 = y0 + 1;
                const float fx1 = xs - x0f, fx0 = 1.0f - fx1;
                const float fy1 = ys - y0f, fy0 = 1.0f - fy1;
                const float ix0 = (x0 >= 0 && x0 < Wl) ? valid : 0.0f;  // fold validity once
                const float ix1 = (x1 >= 0 && x1 < Wl) ? valid : 0.0f;
                const float iy0 = (y0 >= 0 && y0 < Hl) ? 1.0f : 0.0f;
                const float iy1 = (y1 >= 0 && y1 < Hl) ? 1.0f : 0.0f;
                const int xc0 = min(max(x0, 0), Wl - 1);
                const int xc1 = min(max(x1, 0), Wl - 1);
                const int yc0 = min(max(y0, 0), Hl - 1);
                const int yc1 = min(max(y1, 0), Hl - 1);
                const int base = (b * V_ + v) * C_ * HWl;
                s_idx[l][slot][0] = base + yc0 * Wl + xc0;
                s_idx[l][slot][1] = base + yc0 * Wl + xc1;
                s_idx[l][slot][2] = base + yc1 * Wl + xc0;
                s_idx[l][slot][3] = base + yc1 * Wl + xc1;
                s_w[l][slot][0] = fx0 * fy0 * ix0 * iy0;
                s_w[l][slot][1] = fx1 * fy0 * ix1 * iy0;
                s_w[l][slot][2] = fx0 * fy1 * ix0 * iy1;
                s_w[l][slot][3] = fx1 * fy1 * ix1 * iy1;
            }
        };

        v8f czero;
        #pragma unroll
        for (int j = 0; j < 8; ++j) czero[j] = 0.0f;

        const int rowadd = hi ? 8 : 0;

        // ---- WMMA #1: views 0..4 (rows 0..14) ----
        v2f a1 = makeA(0, 15);
        v8f d1 = __builtin_amdgcn_wmma_f32_16x16x4_f32(
            false, a1, false, bfrag, (short)0, czero, false, false);
        #pragma unroll
        for (int j = 0; j < 8; ++j) s_tile[(j + rowadd) * 16 + r] = d1[j];
        __builtin_amdgcn_wave_barrier();
        if (L < 20) {
            const int v = L >> 2, kp = L & 3;
            record(L, v,
                   s_tile[(3 * v + 0) * 16 + kp],
                   s_tile[(3 * v + 1) * 16 + kp],
                   s_tile[(3 * v + 2) * 16 + kp]);
        }
        __builtin_amdgcn_wave_barrier();

        // ---- WMMA #2: view 5 (rows 0..2) ----
        v2f a2 = makeA(5, 3);
        v8f d2 = __builtin_amdgcn_wmma_f32_16x16x4_f32(
            false, a2, false, bfrag, (short)0, czero, false, false);
        #pragma unroll
        for (int j = 0; j < 8; ++j) s_tile[(j + rowadd) * 16 + r] = d2[j];
        __builtin_amdgcn_wave_barrier();
        if (L >= 20 && L < 24) {
            const int kp = L - 20;
            record(L, 5, s_tile[0 * 16 + kp], s_tile[1 * 16 + kp], s_tile[2 * 16 + kp]);
        }
        __builtin_amdgcn_wave_barrier();

        if (L == 0) {
            float sum = 0.0f;
            #pragma unroll
            for (int i = 0; i < NS; ++i) sum += s_valid[i];
            s_scale = 0.5f / fmaxf(sum, 1.0f);
        }
    }
    __syncthreads();

    // ---- gather + weighted accumulation: one channel per thread ----
    const int c = t;
    const int coff0 = c * HW0_;
    const int coff1 = c * HW1_;
    float acc = 0.0f;
    #pragma unroll 4
    for (int s = 0; s < NS; ++s) {
        const int i0 = s_idx[0][s][0] + coff0;
        const int i1 = s_idx[0][s][1] + coff0;
        const int i2 = s_idx[0][s][2] + coff0;
        const int i3 = s_idx[0][s][3] + coff0;
        acc = fmaf(s_w[0][s][0], feat0[i0], acc);
        acc = fmaf(s_w[0][s][1], feat0[i1], acc);
        acc = fmaf(s_w[0][s][2], feat0[i2], acc);
        acc = fmaf(s_w[0][s][3], feat0[i3], acc);
    }
    #pragma unroll 4
    for (int s = 0; s < NS; ++s) {
        const int i0 = s_idx[1][s][0] + coff1;
        const int i1 = s_idx[1][s][1] + coff1;
        const int i2 = s_idx[1][s][2] + coff1;
        const int i3 = s_idx[1][s][3] + coff1;
        acc = fmaf(s_w[1][s][0], feat1[i0], acc);
        acc = fmaf(s_w[1][s][1], feat1[i1], acc);
        acc = fmaf(s_w[1][s][2], feat1[i2], acc);
        acc = fmaf(s_w[1][s][3], feat1[i3], acc);
    }
    out[(size_t)(b * Q_ + q) * C_ + c] = acc * s_scale;
}

extern "C" void kernel_launch(void* const* d_in, const int* in_sizes, int n_in,
                              void* d_out, int out_size, void* d_ws, size_t ws_size,
                              hipStream_t stream) {
    (void)in_sizes; (void)n_in; (void)out_size; (void)d_ws; (void)ws_size;
    const float* feat0 = (const float*)d_in[0];
    const float* feat1 = (const float*)d_in[1];
    const float* refs  = (const float*)d_in[2];
    const float* intr  = (const float*)d_in[3];
    const float* extr  = (const float*)d_in[4];
    float* outp = (float*)d_out;

    dim3 grid(B_ * Q_);   // b-major: consecutive blocks share the L2-resident batch slice
    dim3 block(256);
    hipLaunchKernelGGL(cvss_kernel, grid, block, 0, stream,
                       feat0, feat1, refs, intr, extr, outp);
}